// variant2_5970004542118
// MI455X (gfx1250) — compile-verified
//
#include <hip/hip_runtime.h>
#include <math.h>

typedef float v2f __attribute__((ext_vector_type(2)));
typedef float v8f __attribute__((ext_vector_type(8)));

#define HDIM 128
#define FE   32
#define HS   130   // LDS row stride (floats) to avoid bank conflicts

// ---------------- WMMA helper: one 16x16x4 fp32 step ----------------
__device__ __forceinline__ v8f wmma4(v2f a, v2f b, v8f c) {
    return __builtin_amdgcn_wmma_f32_16x16x4_f32(
        /*neg_a=*/false, a, /*neg_b=*/false, b,
        /*c_mod=*/(short)0, c, /*reuse_a=*/false, /*reuse_b=*/false);
}

// ---------------- init: deg=1 (self loop), cnts=0, pooled=0 ----------------
__global__ void k_init(float* deg, float* cnts, float* pooled, int n, int g) {
    int i = blockIdx.x * blockDim.x + threadIdx.x;
    if (i < n)        deg[i] = 1.0f;
    if (i < g)        cnts[i] = 0.0f;
    if (i < g * HDIM) pooled[i] = 0.0f;
}

__global__ void k_deg(const int* __restrict__ col, float* deg, int e) {
    int i = blockIdx.x * blockDim.x + threadIdx.x;
    if (i < e) atomicAdd(&deg[col[i]], 1.0f);
}

__global__ void k_cnt(const int* __restrict__ batch, float* cnts, int n) {
    int i = blockIdx.x * blockDim.x + threadIdx.x;
    if (i < n) atomicAdd(&cnts[batch[i]], 1.0f);
}

__global__ void k_rsqrt(float* dinv, int n) {
    int i = blockIdx.x * blockDim.x + threadIdx.x;
    if (i < n) { float d = dinv[i]; dinv[i] = d > 0.f ? rsqrtf(d) : 0.f; }
}

// ---------------- Y[nrows x128] = X[nrows x128] @ W[128x128] (fp32 WMMA) ----
// block = 256 threads (8 waves); wave w -> column tile n0 = 16*w.
// Block handles 64 rows = 4 row-tiles per wave; B fragment reused across the
// 4 tiles (4 back-to-back v_wmma per k-step, 4x less weight traffic).
__global__ __launch_bounds__(256)
void k_gemm128(const float* __restrict__ X, const float* __restrict__ W,
               float* __restrict__ Y, int nrows) {
    const int wave = threadIdx.x >> 5;
    const int lane = threadIdx.x & 31;
    const int half = lane >> 4;          // 0: lanes 0-15, 1: lanes 16-31
    const int l    = lane & 15;
    const int m0   = blockIdx.x * 64;
    const int n0   = wave * 16;

    const float* xrow[4];
    #pragma unroll
    for (int t = 0; t < 4; ++t) {
        int mr = m0 + t * 16 + l;
        if (mr > nrows - 1) mr = nrows - 1;       // clamp tail reads
        xrow[t] = X + (size_t)mr * HDIM;
    }
    v8f acc[4];
    #pragma unroll
    for (int t = 0; t < 4; ++t) acc[t] = (v8f){0.f,0.f,0.f,0.f,0.f,0.f,0.f,0.f};

    #pragma unroll 2
    for (int k = 0; k < HDIM; k += 4) {
        v2f b;
        b.x = W[(size_t)(k + half * 2 + 0) * HDIM + n0 + l];
        b.y = W[(size_t)(k + half * 2 + 1) * HDIM + n0 + l];
        #pragma unroll
        for (int t = 0; t < 4; ++t) {
            v2f a;
            a.x = xrow[t][k + half * 2 + 0];
            a.y = xrow[t][k + half * 2 + 1];
            acc[t] = wmma4(a, b, acc[t]);
        }
    }

    if (m0 + 64 <= nrows) {                       // uniform fast path
        #pragma unroll
        for (int t = 0; t < 4; ++t)
            #pragma unroll
            for (int j = 0; j < 8; ++j)
                Y[(size_t)(m0 + t * 16 + j + half * 8) * HDIM + n0 + l] = acc[t][j];
    } else {                                      // tail block
        #pragma unroll
        for (int t = 0; t < 4; ++t)
            #pragma unroll
            for (int j = 0; j < 8; ++j) {
                int m = m0 + t * 16 + j + half * 8;
                if (m < nrows) Y[(size_t)m * HDIM + n0 + l] = acc[t][j];
            }
    }
}

// ---------------- conv helpers ----------------
__global__ void k_selfloop(const float* __restrict__ XW, const float* __restrict__ dinv,
                           float* __restrict__ B, int n) {
    long long idx = (long long)blockIdx.x * blockDim.x + threadIdx.x;
    if (idx < (long long)n * HDIM) {
        int i = (int)(idx >> 7);
        float d = dinv[i];
        B[idx] = XW[idx] * d * d;
    }
}

__global__ void k_msg(const float* __restrict__ XW, const int* __restrict__ row,
                      const int* __restrict__ col, const float* __restrict__ dinv,
                      float* __restrict__ B, int e) {
    long long idx = (long long)blockIdx.x * blockDim.x + threadIdx.x;
    if (idx < (long long)e * HDIM) {
        int ei = (int)(idx >> 7);
        int f  = (int)(idx & (HDIM - 1));
        int r = row[ei], c = col[ei];
        float v = XW[(size_t)r * HDIM + f] * dinv[r] * dinv[c];
        atomicAdd(&B[(size_t)c * HDIM + f], v);
    }
}

__global__ void k_bias_elu(const float* __restrict__ B, const float* __restrict__ bias,
                           float* __restrict__ H, int n) {
    long long idx = (long long)blockIdx.x * blockDim.x + threadIdx.x;
    if (idx < (long long)n * HDIM) {
        int f = (int)(idx & (HDIM - 1));
        float v = B[idx] + bias[f];
        H[idx] = v > 0.f ? v : expm1f(v);
    }
}

// ---------------- attention logits: one wave per (E+N) edge ----------------
__global__ void k_logits(const float* __restrict__ Hb, const int* __restrict__ row,
                         const int* __restrict__ col, const float* __restrict__ Watt,
                         const float* __restrict__ batt, float* __restrict__ logits,
                         int e, int total) {
    int wave = threadIdx.x >> 5, lane = threadIdx.x & 31;
    int i = blockIdx.x * 8 + wave;
    if (i >= total) return;
    int r = (i < e) ? row[i] : (i - e);
    int c = (i < e) ? col[i] : (i - e);
    const float* hr = Hb + (size_t)r * HDIM;
    const float* hc = Hb + (size_t)c * HDIM;
    float p = 0.f;
    #pragma unroll
    for (int t = 0; t < 4; ++t) {
        int f = lane * 4 + t;
        p += hr[f] * Watt[f] + hc[f] * Watt[HDIM + f];
    }
    #pragma unroll
    for (int off = 16; off; off >>= 1) p += __shfl_xor(p, off, 32);
    if (lane == 0) logits[i] = p + batt[0];
}

// ---------------- deterministic two-stage reductions ----------------
__global__ void k_max1(const float* __restrict__ x, int n, float* __restrict__ part) {
    __shared__ float sm[256];
    float m = -INFINITY;
    for (int i = blockIdx.x * 256 + threadIdx.x; i < n; i += gridDim.x * 256)
        m = fmaxf(m, x[i]);
    sm[threadIdx.x] = m; __syncthreads();
    for (int s = 128; s; s >>= 1) {
        if (threadIdx.x < s) sm[threadIdx.x] = fmaxf(sm[threadIdx.x], sm[threadIdx.x + s]);
        __syncthreads();
    }
    if (threadIdx.x == 0) part[blockIdx.x] = sm[0];
}
__global__ void k_max2(const float* __restrict__ part, int p, float* __restrict__ out) {
    __shared__ float sm[256];
    float m = -INFINITY;
    for (int i = threadIdx.x; i < p; i += 256) m = fmaxf(m, part[i]);
    sm[threadIdx.x] = m; __syncthreads();
    for (int s = 128; s; s >>= 1) {
        if (threadIdx.x < s) sm[threadIdx.x] = fmaxf(sm[threadIdx.x], sm[threadIdx.x + s]);
        __syncthreads();
    }
    if (threadIdx.x == 0) out[0] = sm[0];
}
__global__ void k_esum1(const float* __restrict__ x, int n, const float* __restrict__ smax,
                        float* __restrict__ part) {
    __shared__ float sm[256];
    float mx = smax[0], s = 0.f;
    for (int i = blockIdx.x * 256 + threadIdx.x; i < n; i += gridDim.x * 256)
        s += expf(x[i] - mx);
    sm[threadIdx.x] = s; __syncthreads();
    for (int t = 128; t; t >>= 1) {
        if (threadIdx.x < t) sm[threadIdx.x] += sm[threadIdx.x + t];
        __syncthreads();
    }
    if (threadIdx.x == 0) part[blockIdx.x] = sm[0];
}
__global__ void k_esum2(const float* __restrict__ part, int p, float* __restrict__ out) {
    __shared__ float sm[256];
    float s = 0.f;
    for (int i = threadIdx.x; i < p; i += 256) s += part[i];
    sm[threadIdx.x] = s; __syncthreads();
    for (int t = 128; t; t >>= 1) {
        if (threadIdx.x < t) sm[threadIdx.x] += sm[threadIdx.x + t];
        __syncthreads();
    }
    if (threadIdx.x == 0) out[0] = sm[0];
}
__global__ void k_att(float* __restrict__ logits, int e, const float* __restrict__ smax,
                      const float* __restrict__ ssum) {
    int i = blockIdx.x * blockDim.x + threadIdx.x;
    if (i < e) logits[i] = expf(logits[i] - smax[0]) / ssum[0];
}

// ---------------- fused edge MLP + attention-weighted scatter (WMMA) -------
// block = 256 threads (8 waves) handles 64 edges (4 row-tiles per wave);
// wave w -> cols 16*w..16*w+15. B fragments reused across the 4 tiles.
__global__ __launch_bounds__(256)
void k_edge_mlp(const float* __restrict__ EA, const float* __restrict__ We1,
                const float* __restrict__ be1, const float* __restrict__ We2,
                const float* __restrict__ be2, const float* __restrict__ att,
                const int* __restrict__ row, float* __restrict__ Hb, int e) {
    __shared__ float hid[64 * HS];
    const int wave = threadIdx.x >> 5;
    const int lane = threadIdx.x & 31;
    const int half = lane >> 4;
    const int l    = lane & 15;
    const int e0   = blockIdx.x * 64;
    const int n0   = wave * 16;

    // stage 1: hidden = relu(EA[64x32] @ We1[32x128] + be1)
    const float* arow[4];
    #pragma unroll
    for (int t = 0; t < 4; ++t) {
        int er = e0 + t * 16 + l;
        if (er > e - 1) er = e - 1;               // clamp tail reads
        arow[t] = EA + (size_t)er * FE;
    }
    v8f acc[4];
    #pragma unroll
    for (int t = 0; t < 4; ++t) acc[t] = (v8f){0.f,0.f,0.f,0.f,0.f,0.f,0.f,0.f};
    #pragma unroll
    for (int k = 0; k < FE; k += 4) {
        v2f b;
        b.x = We1[(size_t)(k + half * 2 + 0) * HDIM + n0 + l];
        b.y = We1[(size_t)(k + half * 2 + 1) * HDIM + n0 + l];
        #pragma unroll
        for (int t = 0; t < 4; ++t) {
            v2f a;
            a.x = arow[t][k + half * 2 + 0];
            a.y = arow[t][k + half * 2 + 1];
            acc[t] = wmma4(a, b, acc[t]);
        }
    }
    float bv1 = be1[n0 + l];
    #pragma unroll
    for (int t = 0; t < 4; ++t)
        #pragma unroll
        for (int j = 0; j < 8; ++j) {
            int m = t * 16 + j + half * 8;
            float v = acc[t][j] + bv1;
            hid[m * HS + n0 + l] = v > 0.f ? v : 0.f;
        }
    __syncthreads();

    // stage 2: ea = hidden[64x128] @ We2[128x128] + be2 ; scatter att*ea
    v8f acc2[4];
    #pragma unroll
    for (int t = 0; t < 4; ++t) acc2[t] = (v8f){0.f,0.f,0.f,0.f,0.f,0.f,0.f,0.f};
    #pragma unroll 2
    for (int k = 0; k < HDIM; k += 4) {
        v2f b;
        b.x = We2[(size_t)(k + half * 2 + 0) * HDIM + n0 + l];
        b.y = We2[(size_t)(k + half * 2 + 1) * HDIM + n0 + l];
        #pragma unroll
        for (int t = 0; t < 4; ++t) {
            v2f a;
            a.x = hid[(t * 16 + l) * HS + k + half * 2 + 0];
            a.y = hid[(t * 16 + l) * HS + k + half * 2 + 1];
            acc2[t] = wmma4(a, b, acc2[t]);
        }
    }
    float bv2 = be2[n0 + l];
    if (e0 + 64 <= e) {                           // uniform fast path
        #pragma unroll
        for (int t = 0; t < 4; ++t)
            #pragma unroll
            for (int j = 0; j < 8; ++j) {
                int m = e0 + t * 16 + j + half * 8;
                float v = (acc2[t][j] + bv2) * att[m];
                atomicAdd(&Hb[(size_t)row[m] * HDIM + n0 + l], v);
            }
    } else {                                      // tail block
        #pragma unroll
        for (int t = 0; t < 4; ++t)
            #pragma unroll
            for (int j = 0; j < 8; ++j) {
                int m = e0 + t * 16 + j + half * 8;
                if (m < e) {
                    float v = (acc2[t][j] + bv2) * att[m];
                    atomicAdd(&Hb[(size_t)row[m] * HDIM + n0 + l], v);
                }
            }
    }
}

// ---------------- pooling + final FC ----------------
__global__ void k_pool(const float* __restrict__ H2, const int* __restrict__ batch,
                       float* __restrict__ pooled, int n) {
    long long idx = (long long)blockIdx.x * blockDim.x + threadIdx.x;
    if (idx < (long long)n * HDIM) {
        int i = (int)(idx >> 7);
        int f = (int)(idx & (HDIM - 1));
        atomicAdd(&pooled[(size_t)batch[i] * HDIM + f], H2[idx]);
    }
}

__global__ void k_final(const float* __restrict__ pooled, const float* __restrict__ cnts,
                        const float* __restrict__ Wfc, const float* __restrict__ bfc,
                        float* __restrict__ out) {
    __shared__ float sm[HDIM];
    int g = blockIdx.x, f = threadIdx.x;
    sm[f] = pooled[(size_t)g * HDIM + f] * Wfc[f];
    __syncthreads();
    for (int s = 64; s; s >>= 1) {
        if (f < s) sm[f] += sm[f + s];
        __syncthreads();
    }
    if (f == 0) out[g] = sm[0] / fmaxf(cnts[g], 1.0f) + bfc[0];
}

// =====================================================================
extern "C" void kernel_launch(void* const* d_in, const int* in_sizes, int n_in,
                              void* d_out, int out_size, void* d_ws, size_t ws_size,
                              hipStream_t stream) {
    const float* x    = (const float*)d_in[0];
    const int*   ei   = (const int*)  d_in[1];
    const float* ea   = (const float*)d_in[2];
    const int*   bat  = (const int*)  d_in[3];
    const float* W1   = (const float*)d_in[4];
    const float* b1   = (const float*)d_in[5];
    const float* W2   = (const float*)d_in[6];
    const float* b2   = (const float*)d_in[7];
    const float* We1  = (const float*)d_in[8];
    const float* be1  = (const float*)d_in[9];
    const float* We2  = (const float*)d_in[10];
    const float* be2  = (const float*)d_in[11];
    const float* Watt = (const float*)d_in[12];
    const float* batt = (const float*)d_in[13];
    const float* Wfc  = (const float*)d_in[14];
    const float* bfc  = (const float*)d_in[15];

    const int N = in_sizes[0] / HDIM;
    const int E = in_sizes[2] / FE;
    const int G = out_size;           // OUT == 1
    const int T = E + N;              // edges incl. self loops
    const int* row = ei;
    const int* col = ei + E;

    // workspace layout (floats)
    float* A      = (float*)d_ws;                 // N*128  (xw / xw2 / h2)
    float* B      = A + (size_t)N * HDIM;         // N*128  (conv accumulator)
    float* Hb     = B + (size_t)N * HDIM;         // N*128  (h, updated in place)
    float* dinv   = Hb + (size_t)N * HDIM;        // N      (deg -> dinv)
    float* logits = dinv + N;                     // E+N    (logits -> att)
    float* part   = logits + T;                   // 1024
    float* smax   = part + 1024;                  // 1
    float* ssum   = smax + 1;                     // 1
    float* pooled = ssum + 1;                     // G*128
    float* cnts   = pooled + (size_t)G * HDIM;    // G
    (void)ws_size; (void)n_in;

    float* outp = (float*)d_out;
    const int TB = 256;
    const int nBlkNH = (int)(((long long)N * HDIM + TB - 1) / TB);
    const int nBlkEH = (int)(((long long)E * HDIM + TB - 1) / TB);

    // degrees / counts
    k_init  <<<(N + TB - 1) / TB, TB, 0, stream>>>(dinv, cnts, pooled, N, G);
    k_deg   <<<(E + TB - 1) / TB, TB, 0, stream>>>(col, dinv, E);
    k_cnt   <<<(N + TB - 1) / TB, TB, 0, stream>>>(bat, cnts, N);
    k_rsqrt <<<(N + TB - 1) / TB, TB, 0, stream>>>(dinv, N);

    // conv1: h = elu(D^-1/2 A D^-1/2 (x@W1) + b1)
    k_gemm128  <<<(N + 63) / 64, TB, 0, stream>>>(x, W1, A, N);
    k_selfloop <<<nBlkNH, TB, 0, stream>>>(A, dinv, B, N);
    k_msg      <<<nBlkEH, TB, 0, stream>>>(A, row, col, dinv, B, E);
    k_bias_elu <<<nBlkNH, TB, 0, stream>>>(B, b1, Hb, N);

    // global softmax attention over E+N edges
    k_logits <<<(T + 7) / 8, TB, 0, stream>>>(Hb, row, col, Watt, batt, logits, E, T);
    k_max1   <<<1024, TB, 0, stream>>>(logits, T, part);
    k_max2   <<<1, TB, 0, stream>>>(part, 1024, smax);
    k_esum1  <<<1024, TB, 0, stream>>>(logits, T, smax, part);
    k_esum2  <<<1, TB, 0, stream>>>(part, 1024, ssum);
    k_att    <<<(E + TB - 1) / TB, TB, 0, stream>>>(logits, E, smax, ssum);

    // fused edge MLP + weighted scatter: h += segment_sum(att * ea, row)
    k_edge_mlp <<<(E + 63) / 64, TB, 0, stream>>>(ea, We1, be1, We2, be2,
                                                  logits, row, Hb, E);

    // conv2: h2 = elu(D^-1/2 A D^-1/2 (h@W2) + b2)
    k_gemm128  <<<(N + 63) / 64, TB, 0, stream>>>(Hb, W2, A, N);
    k_selfloop <<<nBlkNH, TB, 0, stream>>>(A, dinv, B, N);
    k_msg      <<<nBlkEH, TB, 0, stream>>>(A, row, col, dinv, B, E);
    k_bias_elu <<<nBlkNH, TB, 0, stream>>>(B, b2, A, N);   // h2 -> A

    // mean pool + fc
    k_pool  <<<nBlkNH, TB, 0, stream>>>(A, bat, pooled, N);
    k_final <<<G, HDIM, 0, stream>>>(pooled, cnts, Wfc, bfc, outp);
}